// Yolov3Loss_21835613733241
// MI455X (gfx1250) — compile-verified
//
#include <hip/hip_runtime.h>
#include <stdint.h>

#define TPB 256

typedef unsigned int u32;
typedef u32 u32x4 __attribute__((ext_vector_type(4)));
typedef int  i32x4 __attribute__((ext_vector_type(4)));
typedef int  i32x8 __attribute__((ext_vector_type(8)));

__device__ __forceinline__ float sigm(float x) { return 1.0f / (1.0f + expf(-x)); }

// ---------------------------------------------------------------------------
// One kernel per scale. Block = 256 threads = 8 wave32; grid = (ceil(3*S*S/256), B).
// Each thread owns one (anchor, row, col) cell and streams its 169 channel
// values once. boxes_target[b] (800 B) is DMA'd into LDS by the Tensor Data
// Mover (CDNA5 TDM) and reused by every cell of the block.
// ---------------------------------------------------------------------------
template <int S>
__global__ __launch_bounds__(TPB) void yolo_scale_kernel(
    const float* __restrict__ out,    // (B, 255, S, S)
    const float* __restrict__ loc,    // (B, 3, 4, S, S)
    const float* __restrict__ cls,    // (B, 3, 80, S, S)
    const float* __restrict__ boxes,  // (B, 50, 4)
    float aw0, float ah0, float aw1, float ah1, float aw2, float ah2,
    float* __restrict__ partials)     // (B*gridDim.x, 4): ll, lconf, lcls, npos
{
  constexpr int SS = S * S;
  constexpr int N  = 3 * SS;

  __shared__ float sboxes[200];
  __shared__ float sred[TPB / 32][4];

  const int b = blockIdx.y;
  const float* gb = boxes + (size_t)b * 200;

#if defined(__gfx1250__) && __has_builtin(__builtin_amdgcn_tensor_load_to_lds)
  if (threadIdx.x == 0) {  // TENSOR ops ignore EXEC -> exactly one DMA per block
    unsigned long long ga = (unsigned long long)(uintptr_t)gb;
    u32 lds_addr = (u32)(uintptr_t)(void*)sboxes;  // low 32 bits = LDS byte addr
    u32x4 g0;
    g0.x = 1u;                                              // count=1 (valid D#)
    g0.y = lds_addr;                                        // lds_addr
    g0.z = (u32)ga;                                         // global_addr[31:0]
    g0.w = ((u32)(ga >> 32) & 0x01FFFFFFu) | (2u << 30);    // addr[56:32] | type=2
    i32x8 g1;
    g1[0] = (int)(2u << 16);    // data_size = 4 bytes
    g1[1] = (int)(200u << 16);  // tensor_dim0 = 200 elements
    g1[2] = (int)(1u << 16);    // tensor_dim0 hi=0, tensor_dim1 = 1
    g1[3] = (int)(200u << 16);  // tensor_dim1 hi=0, tile_dim0 = 200
    g1[4] = 1;                  // tile_dim1 = 1, tile_dim2 = 0
    g1[5] = 200;                // tensor_dim0_stride = 200
    g1[6] = 0;
    g1[7] = 0;
    i32x4 z4 = {0, 0, 0, 0};
#if __clang_major__ >= 23
    i32x8 z8 = {0, 0, 0, 0, 0, 0, 0, 0};
    __builtin_amdgcn_tensor_load_to_lds(g0, g1, z4, z4, z8, 0);
#else
    __builtin_amdgcn_tensor_load_to_lds(g0, g1, z4, z4, 0);
#endif
#if __has_builtin(__builtin_amdgcn_s_wait_tensorcnt)
    __builtin_amdgcn_s_wait_tensorcnt(0);
#else
    asm volatile("s_wait_tensorcnt 0" ::: "memory");
#endif
  }
#else
  if (threadIdx.x < 200) sboxes[threadIdx.x] = gb[threadIdx.x];
#endif
  __syncthreads();

  float ll = 0.0f, lcf = 0.0f, lkf = 0.0f, pcnt = 0.0f;

  const int t = blockIdx.x * TPB + threadIdx.x;
  if (t < N) {
    const int a  = t / SS;
    const int r  = t - a * SS;
    const int gi = r / S;        // row (y)
    const int gj = r - gi * S;   // col (x)

    const float* op = out + ((size_t)b * 255 + (size_t)a * 85) * SS + r;
    const float* cp = cls + (((size_t)b * 3 + a) * 80) * SS + r;
    const float* lp = loc + (((size_t)b * 3 + a) * 4) * SS + r;

    // ---- positive-cell mask: max over 80 class-target channels > 0 ----
    float cmax = -3.4e38f;
#pragma unroll 8
    for (int c = 0; c < 80; ++c) cmax = fmaxf(cmax, cp[(size_t)c * SS]);
    const bool pos = cmax > 0.0f;
    pcnt = pos ? 1.0f : 0.0f;

    // ---- loc loss on raw tx,ty,tw,th ----
    const float tx = op[0];
    const float ty = op[SS];
    const float tw = op[2 * (size_t)SS];
    const float th = op[3 * (size_t)SS];
    const float tc = op[4 * (size_t)SS];
    if (pos) {
      float d0 = tx - lp[0];
      float d1 = ty - lp[SS];
      float d2 = tw - lp[2 * (size_t)SS];
      float d3 = th - lp[3 * (size_t)SS];
      ll = d0 * d0 + d1 * d1 + d2 * d2 + d3 * d3;
    }

    // ---- decode box, max IoU vs 50 LDS targets ----
    const float bx = sigm(tx) + (float)gj;
    const float by = sigm(ty) + (float)gi;
    const float aw = (a == 0) ? aw0 : ((a == 1) ? aw1 : aw2);
    const float ah = (a == 0) ? ah0 : ((a == 1) ? ah1 : ah2);
    const float bw = expf(tw) * aw;
    const float bh = expf(th) * ah;
    const float x1 = bx - 0.5f * bw, y1 = by - 0.5f * bh;
    const float x2 = bx + 0.5f * bw, y2 = by + 0.5f * bh;
    const float areaA = (x2 - x1) * (y2 - y1);
    float ct = 0.0f;  // IoU >= 0 always
#pragma unroll 5
    for (int m = 0; m < 50; ++m) {
      float gx1 = sboxes[4 * m + 0], gy1 = sboxes[4 * m + 1];
      float gx2 = sboxes[4 * m + 2], gy2 = sboxes[4 * m + 3];
      float iw = fmaxf(fminf(x2, gx2) - fmaxf(x1, gx1), 0.0f);
      float ih = fmaxf(fminf(y2, gy2) - fmaxf(y1, gy1), 0.0f);
      float inter = iw * ih;
      float areaB = (gx2 - gx1) * (gy2 - gy1);
      ct = fmaxf(ct, inter / (areaA + areaB - inter));
    }
    const float cpred = sigm(tc);
    const float msk = pos ? 1.0f : 0.1f;
    const float dc = msk * (cpred - ct);
    lcf = dc * dc;

    // ---- stable softmax over 80 logits kept in VGPRs ----
    float lg[80];
#pragma unroll
    for (int c = 0; c < 80; ++c) lg[c] = op[(size_t)(5 + c) * SS];
    float mx = lg[0];
#pragma unroll
    for (int c = 1; c < 80; ++c) mx = fmaxf(mx, lg[c]);
    float se = 0.0f;
#pragma unroll
    for (int c = 0; c < 80; ++c) { float e = expf(lg[c] - mx); lg[c] = e; se += e; }
    const float inv = 1.0f / se;
#pragma unroll 8
    for (int c = 0; c < 80; ++c) {
      float cv = cp[(size_t)c * SS];
      if (cv > 0.0f) { float d = lg[c] * inv - cv; lkf += d * d; }
    }
  }

  // ---- deterministic block reduction: shuffle within wave32, LDS across ----
  for (int o = 16; o > 0; o >>= 1) {
    ll   += __shfl_down(ll, o, 32);
    lcf  += __shfl_down(lcf, o, 32);
    lkf  += __shfl_down(lkf, o, 32);
    pcnt += __shfl_down(pcnt, o, 32);
  }
  const int lane = threadIdx.x & 31;
  const int wv   = threadIdx.x >> 5;
  if (lane == 0) {
    sred[wv][0] = ll; sred[wv][1] = lcf; sred[wv][2] = lkf; sred[wv][3] = pcnt;
  }
  __syncthreads();
  if (threadIdx.x == 0) {
    float a0 = 0, a1 = 0, a2 = 0, a3 = 0;
    for (int w = 0; w < TPB / 32; ++w) {
      a0 += sred[w][0]; a1 += sred[w][1]; a2 += sred[w][2]; a3 += sred[w][3];
    }
    float* p = partials + ((size_t)blockIdx.y * gridDim.x + blockIdx.x) * 4;
    p[0] = a0; p[1] = a1; p[2] = a2; p[3] = a3;
  }
}

// ---------------------------------------------------------------------------
// Single-block deterministic finalize: sum all per-block partials (pos count
// only from scale 0), combine: loss = 5*(ll+lc+lk) / B / num_pos.
// ---------------------------------------------------------------------------
__global__ __launch_bounds__(TPB) void yolo_finalize(
    const float* __restrict__ parts, int n0, int n1, int n2, int B,
    float* __restrict__ outv)
{
  __shared__ float s[4][TPB];
  const int tid = threadIdx.x;
  const int total = n0 + n1 + n2;
  float ll = 0, lc = 0, lk = 0, pc = 0;
  for (int idx = tid; idx < total; idx += TPB) {
    const float* p = parts + (size_t)idx * 4;
    ll += p[0]; lc += p[1]; lk += p[2];
    if (idx < n0) pc += p[3];
  }
  s[0][tid] = ll; s[1][tid] = lc; s[2][tid] = lk; s[3][tid] = pc;
  __syncthreads();
  for (int o = TPB / 2; o > 0; o >>= 1) {
    if (tid < o) {
      s[0][tid] += s[0][tid + o];
      s[1][tid] += s[1][tid + o];
      s[2][tid] += s[2][tid + o];
      s[3][tid] += s[3][tid + o];
    }
    __syncthreads();
  }
  if (tid == 0) {
    float num = 5.0f * (s[0][0] + s[1][0] + s[2][0]);
    outv[0] = num / (float)B / s[3][0];
  }
}

// ---------------------------------------------------------------------------
// Host launcher. Input order (setup_inputs dict order):
//   0:output0 1:loc0 2:cls0 3:boxes0  4:output1 5:loc1 6:cls1 7:boxes1
//   8:output2 9:loc2 10:cls2 11:boxes2
// ---------------------------------------------------------------------------
extern "C" void kernel_launch(void* const* d_in, const int* in_sizes, int n_in,
                              void* d_out, int out_size, void* d_ws, size_t ws_size,
                              hipStream_t stream) {
  (void)n_in; (void)out_size; (void)ws_size;
  const float* out0 = (const float*)d_in[0];
  const float* loc0 = (const float*)d_in[1];
  const float* cls0 = (const float*)d_in[2];
  const float* box0 = (const float*)d_in[3];
  const float* out1 = (const float*)d_in[4];
  const float* loc1 = (const float*)d_in[5];
  const float* cls1 = (const float*)d_in[6];
  const float* box1 = (const float*)d_in[7];
  const float* out2 = (const float*)d_in[8];
  const float* loc2 = (const float*)d_in[9];
  const float* cls2 = (const float*)d_in[10];
  const float* box2 = (const float*)d_in[11];

  const int B = in_sizes[0] / (255 * 13 * 13);

  const int gx0 = (3 * 13 * 13 + TPB - 1) / TPB;   // 2
  const int gx1 = (3 * 26 * 26 + TPB - 1) / TPB;   // 8
  const int gx2 = (3 * 52 * 52 + TPB - 1) / TPB;   // 32
  const int n0 = gx0 * B, n1 = gx1 * B, n2 = gx2 * B;

  float* ws = (float*)d_ws;
  float* p0 = ws;
  float* p1 = ws + (size_t)n0 * 4;
  float* p2 = ws + (size_t)(n0 + n1) * 4;

  // anchors scaled to grid units: w/416*S
  const float f0 = 13.0f / 416.0f, f1 = 26.0f / 416.0f, f2 = 52.0f / 416.0f;

  yolo_scale_kernel<13><<<dim3(gx0, B), TPB, 0, stream>>>(
      out0, loc0, cls0, box0,
      116.0f * f0, 90.0f * f0, 156.0f * f0, 198.0f * f0, 373.0f * f0, 326.0f * f0,
      p0);
  yolo_scale_kernel<26><<<dim3(gx1, B), TPB, 0, stream>>>(
      out1, loc1, cls1, box1,
      30.0f * f1, 61.0f * f1, 62.0f * f1, 45.0f * f1, 59.0f * f1, 119.0f * f1,
      p1);
  yolo_scale_kernel<52><<<dim3(gx2, B), TPB, 0, stream>>>(
      out2, loc2, cls2, box2,
      10.0f * f2, 13.0f * f2, 16.0f * f2, 30.0f * f2, 33.0f * f2, 23.0f * f2,
      p2);

  yolo_finalize<<<1, TPB, 0, stream>>>(ws, n0, n1, n2, B, (float*)d_out);
}